// ParametricFIFO_3204045603618
// MI455X (gfx1250) — compile-verified
//
#include <hip/hip_runtime.h>

#define HID   64
#define FF2   128
#define INNER 24
#define SEQN  32
#define CAPN  8
#define BATCH 8192
#define NPOS  17   // positions 14..30 are the only ones consumed

typedef __attribute__((ext_vector_type(2))) float v2f;
typedef __attribute__((ext_vector_type(8))) float v8f;

__device__ __forceinline__ v8f wmma4(v2f a, v2f b, v8f c) {
  // D = A(16x4) * B(4x16) + C, full f32 — matches reference precision.
  return __builtin_amdgcn_wmma_f32_16x16x4_f32(false, a, false, b, (short)0, c, false, false);
}

// Async global->LDS DMA (CDNA5): tracked by ASYNCcnt, no VGPR landing zone.
// vdst = LDS byte offset (low 32 bits of generic shared pointer), vaddr = 64-bit global.
__device__ __forceinline__ void async_copy16(void* lds_dst, const void* g_src) {
  asm volatile("global_load_async_to_lds_b128 %0, %1, off"
               :: "v"((unsigned int)(unsigned long long)(uintptr_t)lds_dst),
                  "v"((unsigned long long)(uintptr_t)g_src)
               : "memory");
}
__device__ __forceinline__ void wait_async0() {
  asm volatile("s_wait_asynccnt 0" ::: "memory");
}

// ---------------------------------------------------------------------------
// Kernel 1: encoder for the 17 needed positions per sample.
// One wave = one 16-row tile of (sample,pos) space. f32 WMMA for both layers.
// Outer n-tile loops are kept ROLLED so live state stays small (<256 VGPRs);
// latency is hidden by the 8-16 resident waves per WGP, not by a giant
// intra-wave software pipeline.
// ---------------------------------------------------------------------------
__global__ __launch_bounds__(128) void enc_kernel(
    const int* __restrict__ seqs, const float* __restrict__ embed,
    const float* __restrict__ ff_w1, const float* __restrict__ ff_b1,
    const float* __restrict__ ff_w2, const float* __restrict__ ff_b2,
    const float* __restrict__ ln_g,  const float* __restrict__ ln_b,
    float* __restrict__ keys, float* __restrict__ vals, float* __restrict__ qout)
{
  __shared__ float s_w1[HID * FF2];          // 32 KB
  __shared__ float s_w2[FF2 * HID];          // 32 KB
  __shared__ float s_h[4][16][HID + 4];      // h rows (f32, residual source)
  __shared__ float s_t1[4][16][FF2 + 4];     // hidden activations
  __shared__ float s_x[4][16][HID + 4];      // x = h + f
  __shared__ float s_stat[4][16][2];         // LN mu / rstd

  const int tid  = threadIdx.x;
  const int wave = tid >> 5;
  const int lane = tid & 31;
  const int tile = blockIdx.x * 4 + wave;    // 8704 tiles total, exact

  // Stage FF weights via async DMA (overlaps with the embedding gather below).
#pragma unroll 1
  for (int i = tid * 4; i < HID * FF2; i += 128 * 4) async_copy16(&s_w1[i], &ff_w1[i]);
#pragma unroll 1
  for (int i = tid * 4; i < FF2 * HID; i += 128 * 4) async_copy16(&s_w2[i], &ff_w2[i]);

#pragma unroll 1
  for (int r = 0; r < 16; ++r) {
    int gr = tile * 16 + r;
    int s  = gr / NPOS;
    int j  = gr - s * NPOS;                  // 0..16 -> pos 14..30
    int tok = seqs[s * SEQN + 14 + j];
    s_h[wave][r][lane]      = embed[tok * HID + lane];
    s_h[wave][r][lane + 32] = embed[tok * HID + lane + 32];
  }
  wait_async0();
  __syncthreads();

  const int arow = lane & 15;   // A-fragment row / B,C column
  const int part = lane >> 4;   // lane half selector

  // A fragments for layer 1 (K=64 -> 16 wmma k-steps), held in VGPRs (32 regs,
  // reused across all 8 n-tiles). Contiguous pairs -> single ds_load_b64 each.
  v2f a1[16];
#pragma unroll
  for (int ks = 0; ks < 16; ++ks)
    a1[ks] = *(const v2f*)&s_h[wave][arow][ks * 4 + part * 2];

  // Layer 1: t1 = relu(h @ W1 + b1), N=128 -> 8 n-tiles (rolled).
#pragma unroll 1
  for (int nt = 0; nt < 8; ++nt) {
    v8f c = {};
#pragma unroll
    for (int ks = 0; ks < 16; ++ks) {
      v2f b;
      b.x = s_w1[(ks * 4 + part * 2 + 0) * FF2 + nt * 16 + arow];
      b.y = s_w1[(ks * 4 + part * 2 + 1) * FF2 + nt * 16 + arow];
      c = wmma4(a1[ks], b, c);
    }
    float bias = ff_b1[nt * 16 + arow];
#pragma unroll
    for (int i = 0; i < 8; ++i) {
      float v = c[i] + bias;
      s_t1[wave][i + 8 * part][nt * 16 + arow] = v > 0.0f ? v : 0.0f;
    }
  }

  // Layer 2: f = t1 @ W2 + b2, N=64 -> 4 n-tiles (rolled), K=128 -> 32 k-steps.
#pragma unroll 1
  for (int nt = 0; nt < 4; ++nt) {
    v8f c = {};
#pragma unroll
    for (int ks = 0; ks < 32; ++ks) {
      v2f a = *(const v2f*)&s_t1[wave][arow][ks * 4 + part * 2];
      v2f b;
      b.x = s_w2[(ks * 4 + part * 2 + 0) * HID + nt * 16 + arow];
      b.y = s_w2[(ks * 4 + part * 2 + 1) * HID + nt * 16 + arow];
      c = wmma4(a, b, c);
    }
    float bias = ff_b2[nt * 16 + arow];
#pragma unroll
    for (int i = 0; i < 8; ++i) {
      int r  = i + 8 * part;
      int cc = nt * 16 + arow;
      s_x[wave][r][cc] = c[i] + bias + s_h[wave][r][cc];
    }
  }

  // LayerNorm: two lanes per row, butterfly-combine partial sums.
  {
    int r = lane >> 1, hh = lane & 1;
    float sum = 0.0f, sq = 0.0f;
#pragma unroll
    for (int c = 0; c < 32; ++c) {
      float v = s_x[wave][r][hh * 32 + c];
      sum += v; sq += v * v;
    }
    sum += __shfl_xor(sum, 1);
    sq  += __shfl_xor(sq, 1);
    float mu  = sum * (1.0f / 64.0f);
    float var = sq * (1.0f / 64.0f) - mu * mu;
    s_stat[wave][r][0] = mu;
    s_stat[wave][r][1] = rsqrtf(var + 1e-5f);
  }

  float g0 = ln_g[lane], g1 = ln_g[lane + 32];
  float bb0 = ln_b[lane], bb1 = ln_b[lane + 32];
#pragma unroll 1
  for (int r = 0; r < 16; ++r) {
    int gr = tile * 16 + r;
    int s  = gr / NPOS;
    int j  = gr - s * NPOS;
    float* dst;
    if (j == 16)            dst = qout + s * HID;                       // pos 30
    else if ((j & 1) == 0)  dst = keys + (s * CAPN + (j >> 1)) * HID;   // 14,16,..,28
    else                    dst = vals + (s * CAPN + (j >> 1)) * HID;   // 15,17,..,29
    float mu = s_stat[wave][r][0], rstd = s_stat[wave][r][1];
    dst[lane]      = g0 * (s_x[wave][r][lane]      - mu) * rstd + bb0;
    dst[lane + 32] = g1 * (s_x[wave][r][lane + 32] - mu) * rstd + bb1;
  }
}

// ---------------------------------------------------------------------------
// Kernel 2: per-sample 8-step Adam training of the 64->24->64 MLP.
// One wave per sample. Params in VGPRs (lane j owns w1 col j; lane l owns
// w2 out-cols l and l+32). Adam m/v in LDS, bank-conflict-free layouts.
// ---------------------------------------------------------------------------
__global__ __launch_bounds__(128) void ttt_kernel(
    const float* __restrict__ keys, const float* __restrict__ vals,
    const float* __restrict__ qin,
    const float* __restrict__ w1_0, const float* __restrict__ b1_0,
    const float* __restrict__ w2_0, const float* __restrict__ b2_0,
    float* __restrict__ hq)
{
  __shared__ float s_kv[4][1088];            // 8 keys + 8 vals + q per sample
  __shared__ float s_w1m[4][HID * INNER];    // layout [k][j]
  __shared__ float s_w1v[4][HID * INNER];
  __shared__ float s_w2m[4][INNER * HID];    // layout [j][o]
  __shared__ float s_w2v[4][INNER * HID];
  __shared__ float s_part[4][INNER * 33];    // padded transpose scratch

  const int wave = threadIdx.x >> 5;
  const int lane = threadIdx.x & 31;
  const int s    = blockIdx.x * 4 + wave;

  float* kv   = s_kv[wave];
  float* w1m  = s_w1m[wave];
  float* w1v  = s_w1v[wave];
  float* w2m  = s_w2m[wave];
  float* w2v  = s_w2v[wave];
  float* part = s_part[wave];

  // Async-stage this sample's keys/vals/q into LDS (overlaps param loads).
#pragma unroll
  for (int i = 0; i < 4; ++i) {
    async_copy16(&kv[(i * 32 + lane) * 4],       &keys[s * 512 + (i * 32 + lane) * 4]);
    async_copy16(&kv[512 + (i * 32 + lane) * 4], &vals[s * 512 + (i * 32 + lane) * 4]);
  }
  if (lane < 16) async_copy16(&kv[1024 + lane * 4], &qin[s * 64 + lane * 4]);

#pragma unroll 1
  for (int i = lane; i < HID * INNER; i += 32) { w1m[i] = 0.f; w1v[i] = 0.f; w2m[i] = 0.f; w2v[i] = 0.f; }

  const int jl = lane < INNER ? lane : INNER - 1;  // lanes 24..31 duplicate col 23 (benign)

  float w1p[HID];
#pragma unroll
  for (int k = 0; k < HID; ++k) w1p[k] = w1_0[k * INNER + jl];
  float b1p = b1_0[jl], b1m = 0.f, b1v = 0.f;
  float w2p0[INNER], w2p1[INNER];
#pragma unroll
  for (int j = 0; j < INNER; ++j) { w2p0[j] = w2_0[j * HID + lane]; w2p1[j] = w2_0[j * HID + lane + 32]; }
  float b2p0 = b2_0[lane], b2p1 = b2_0[lane + 32];
  float b2m0 = 0.f, b2v0 = 0.f, b2m1 = 0.f, b2v1 = 0.f;

  wait_async0();   // kv resident in LDS before first use

  const float B1c = 0.9f, B2c = 0.999f, LRc = 0.05f, EPSc = 1e-8f;
  float p1 = 1.0f, p2 = 1.0f;

#pragma unroll 1
  for (int t = 0; t < CAPN; ++t) {
    p1 *= B1c; p2 *= B2c;
    const float ib1 = 1.0f / (1.0f - p1);   // 1/bias-corr-1
    const float bc2 = 1.0f - p2;            // bias-corr-2

    const float* kh = kv + t * 64;
    const float* vh = kv + 512 + t * 64;

    // forward: a1[jl] = b1 + kh . w1[:,jl]
    float a1 = b1p;
#pragma unroll
    for (int k = 0; k < HID; ++k) a1 = fmaf(kh[k], w1p[k], a1);
    float r = a1 > 0.f ? a1 : 0.f;

    float rr[INNER];
#pragma unroll
    for (int j = 0; j < INNER; ++j) rr[j] = __shfl(r, j);

    float pr0 = b2p0, pr1 = b2p1;
#pragma unroll
    for (int j = 0; j < INNER; ++j) { pr0 = fmaf(rr[j], w2p0[j], pr0); pr1 = fmaf(rr[j], w2p1[j], pr1); }
    float gp0 = (2.0f / 64.0f) * (pr0 - vh[lane]);
    float gp1 = (2.0f / 64.0f) * (pr1 - vh[lane + 32]);

    // per-lane partials of g_r (uses OLD w2) -> LDS transpose
#pragma unroll
    for (int j = 0; j < INNER; ++j)
      part[j * 33 + lane] = w2p0[j] * gp0 + w2p1[j] * gp1;

    // Adam update: w2, b2
#pragma unroll
    for (int j = 0; j < INNER; ++j) {
      int i0 = j * HID + lane;
      float g0 = rr[j] * gp0;
      float m0 = B1c * w2m[i0] + (1.f - B1c) * g0;
      float v0 = B2c * w2v[i0] + (1.f - B2c) * g0 * g0;
      w2m[i0] = m0; w2v[i0] = v0;
      w2p0[j] -= LRc * (m0 * ib1) / (sqrtf(v0 / bc2) + EPSc);
      int i1 = i0 + 32;
      float g1 = rr[j] * gp1;
      float m1 = B1c * w2m[i1] + (1.f - B1c) * g1;
      float v1 = B2c * w2v[i1] + (1.f - B2c) * g1 * g1;
      w2m[i1] = m1; w2v[i1] = v1;
      w2p1[j] -= LRc * (m1 * ib1) / (sqrtf(v1 / bc2) + EPSc);
    }
    b2m0 = B1c * b2m0 + (1.f - B1c) * gp0; b2v0 = B2c * b2v0 + (1.f - B2c) * gp0 * gp0;
    b2p0 -= LRc * (b2m0 * ib1) / (sqrtf(b2v0 / bc2) + EPSc);
    b2m1 = B1c * b2m1 + (1.f - B1c) * gp1; b2v1 = B2c * b2v1 + (1.f - B2c) * gp1 * gp1;
    b2p1 -= LRc * (b2m1 * ib1) / (sqrtf(b2v1 / bc2) + EPSc);

    // reduce g_r: lane jl sums its 32 partials (stride-33 => conflict-free)
    float gr = 0.f;
#pragma unroll
    for (int i = 0; i < 32; ++i) gr += part[jl * 33 + i];
    float ga1 = (a1 > 0.f) ? gr : 0.f;

    b1m = B1c * b1m + (1.f - B1c) * ga1; b1v = B2c * b1v + (1.f - B2c) * ga1 * ga1;
    b1p -= LRc * (b1m * ib1) / (sqrtf(b1v / bc2) + EPSc);

    // Adam update: w1 column jl  (grad = kh[k] * ga1)
#pragma unroll
    for (int k = 0; k < HID; ++k) {
      float g = kh[k] * ga1;
      int idx = k * INNER + jl;
      float m = B1c * w1m[idx] + (1.f - B1c) * g;
      float v = B2c * w1v[idx] + (1.f - B2c) * g * g;
      w1m[idx] = m; w1v[idx] = v;
      w1p[k] -= LRc * (m * ib1) / (sqrtf(v / bc2) + EPSc);
    }
  }

  // final forward on q with trained params
  const float* qv = kv + 1024;
  float a1 = b1p;
#pragma unroll
  for (int k = 0; k < HID; ++k) a1 = fmaf(qv[k], w1p[k], a1);
  float r = a1 > 0.f ? a1 : 0.f;
  float rr[INNER];
#pragma unroll
  for (int j = 0; j < INNER; ++j) rr[j] = __shfl(r, j);
  float pr0 = b2p0, pr1 = b2p1;
#pragma unroll
  for (int j = 0; j < INNER; ++j) { pr0 = fmaf(rr[j], w2p0[j], pr0); pr1 = fmaf(rr[j], w2p1[j], pr1); }
  hq[s * 64 + lane]      = pr0;
  hq[s * 64 + lane + 32] = pr1;
}

// ---------------------------------------------------------------------------
// Kernel 3: out = hq[8192x64] @ out_w[64x64] + out_b  (f32 WMMA)
// ---------------------------------------------------------------------------
__global__ __launch_bounds__(256) void out_kernel(
    const float* __restrict__ hq, const float* __restrict__ out_w,
    const float* __restrict__ out_b, float* __restrict__ out)
{
  const int wave = threadIdx.x >> 5;
  const int lane = threadIdx.x & 31;
  const int tile = blockIdx.x * 8 + wave;    // 512 tiles
  const int arow = lane & 15;
  const int part = lane >> 4;

  v2f a[16];
#pragma unroll
  for (int ks = 0; ks < 16; ++ks) {
    const float* row = hq + (tile * 16 + arow) * 64 + ks * 4 + part * 2;
    a[ks].x = row[0];
    a[ks].y = row[1];
  }
#pragma unroll 1
  for (int nt = 0; nt < 4; ++nt) {
    v8f c = {};
#pragma unroll
    for (int ks = 0; ks < 16; ++ks) {
      v2f b;
      b.x = out_w[(ks * 4 + part * 2 + 0) * 64 + nt * 16 + arow];
      b.y = out_w[(ks * 4 + part * 2 + 1) * 64 + nt * 16 + arow];
      c = wmma4(a[ks], b, c);
    }
    float bias = out_b[nt * 16 + arow];
#pragma unroll
    for (int i = 0; i < 8; ++i)
      out[(tile * 16 + i + 8 * part) * 64 + nt * 16 + arow] = c[i] + bias;
  }
}

// ---------------------------------------------------------------------------
extern "C" void kernel_launch(void* const* d_in, const int* in_sizes, int n_in,
                              void* d_out, int out_size, void* d_ws, size_t ws_size,
                              hipStream_t stream) {
  const int*   seqs   = (const int*)  d_in[0];
  const float* embed  = (const float*)d_in[1];
  const float* ff_w1  = (const float*)d_in[2];
  const float* ff_b1  = (const float*)d_in[3];
  const float* ff_w2  = (const float*)d_in[4];
  const float* ff_b2  = (const float*)d_in[5];
  const float* ln_g   = (const float*)d_in[6];
  const float* ln_b   = (const float*)d_in[7];
  const float* mlp_w1 = (const float*)d_in[8];
  const float* mlp_b1 = (const float*)d_in[9];
  const float* mlp_w2 = (const float*)d_in[10];
  const float* mlp_b2 = (const float*)d_in[11];
  const float* out_w  = (const float*)d_in[12];
  const float* out_b  = (const float*)d_in[13];
  float* out = (float*)d_out;

  float* ws   = (float*)d_ws;                       // ~36 MB used
  float* keys = ws;                                 // [8192][8][64]
  float* vals = keys + (size_t)BATCH * CAPN * HID;  // [8192][8][64]
  float* q    = vals + (size_t)BATCH * CAPN * HID;  // [8192][64]
  float* hq   = q    + (size_t)BATCH * HID;         // [8192][64]

  enc_kernel<<<(BATCH * NPOS / 16) / 4, 128, 0, stream>>>(
      seqs, embed, ff_w1, ff_b1, ff_w2, ff_b2, ln_g, ln_b, keys, vals, q);
  ttt_kernel<<<BATCH / 4, 128, 0, stream>>>(
      keys, vals, q, mlp_w1, mlp_b1, mlp_w2, mlp_b2, hq);
  out_kernel<<<(BATCH / 16) / 8, 256, 0, stream>>>(hq, out_w, out_b, out);
}